// ResidualVectorQuantizer_46205258170719
// MI455X (gfx1250) — compile-verified
//
#include <hip/hip_runtime.h>

// ---------------------------------------------------------------------------
// Residual VQ (Q=8, K=1024, D=8) for gfx1250 using V_WMMA_F32_16X16X4_F32.
// Score s(n) = |c_n|^2 - 2 r.c_n : the -2r.c part is a WMMA chain (C starts
// at inline 0), |c_n|^2 added at compare time.  Argmin across columns is a
// fully branchless packed-u64 butterfly + one ds_bpermute broadcast.
// ---------------------------------------------------------------------------

typedef __attribute__((ext_vector_type(2))) float v2f;
typedef __attribute__((ext_vector_type(8))) float v8f;

#define B_ 32
#define T_ 8192
#define D_ 8
#define Q_ 8
#define K_ 1024
#define BT_ (B_ * T_)          // 262144 tokens
#define CBS 10                 // padded LDS row stride (floats) -> no bank conflicts
#define WAVES 8
#define TPW 16                 // tokens per wave (one WMMA M-tile)
#define TPB (WAVES * TPW)      // 128 tokens per block
#define NWG (BT_ / TPB)        // 2048 workgroups

__device__ __forceinline__ int shflx_i(int v, int mask, int lane) {
    return __builtin_amdgcn_ds_bpermute(((lane ^ mask) << 2), v);
}
__device__ __forceinline__ float shflx_f(float v, int mask, int lane) {
    return __int_as_float(
        __builtin_amdgcn_ds_bpermute(((lane ^ mask) << 2), __float_as_int(v)));
}

// monotone float -> u32 map (ascending); makes argmin a pure integer min
__device__ __forceinline__ unsigned sortable(float f) {
    unsigned b = __float_as_uint(f);
    return b ^ (((unsigned)((int)b >> 31)) | 0x80000000u);
}

__global__ __launch_bounds__(256) void rvq_wmma_kernel(
    const float* __restrict__ x, const float* __restrict__ codebooks,
    const float* __restrict__ post_scale, const float* __restrict__ post_bias,
    const float* __restrict__ conv_w, const float* __restrict__ conv_b,
    float* __restrict__ out_q, int* __restrict__ out_codes,
    float* __restrict__ loss_partials)
{
    __shared__ float cb_sh[K_ * CBS];   // padded codebook rows (reused as qsum stage)
    __shared__ float cn_sh[K_];         // |c|^2 per codeword
    __shared__ float ls_sh[WAVES];

    const int tid  = threadIdx.x;
    const int lane = tid & 31;
    const int wave = tid >> 5;
    const int hl   = lane >> 4;     // half-wave: 0 => dims {0,1,4,5}, 1 => {2,3,6,7}
    const int tl   = lane & 15;     // token within wave / column within N-tile
    const int c0   = hl * 2;
    const int tok  = blockIdx.x * TPB + wave * TPW + tl;

    // Residual registers in exact WMMA f32 16x4 A-matrix layout.
    const float* xr = x + (size_t)tok * D_;
    float r0 = xr[c0 + 0], r1 = xr[c0 + 1], r2 = xr[c0 + 4], r3 = xr[c0 + 5];
    float q0 = 0.f, q1 = 0.f, q2 = 0.f, q3 = 0.f;
    float lacc = 0.f;

    for (int q = 0; q < Q_; ++q) {
        __syncthreads();  // previous step's LDS readers done before overwrite
        const float* cbg = codebooks + (size_t)q * K_ * D_;
        for (int r = tid; r < K_; r += 256) {
            const float4 a = ((const float4*)(cbg + r * D_))[0];
            const float4 b = ((const float4*)(cbg + r * D_))[1];
            float* dst = &cb_sh[r * CBS];
            dst[0] = a.x; dst[1] = a.y; dst[2] = a.z; dst[3] = a.w;
            dst[4] = b.x; dst[5] = b.y; dst[6] = b.z; dst[7] = b.w;
            cn_sh[r] = a.x*a.x + a.y*a.y + a.z*a.z + a.w*a.w
                     + b.x*b.x + b.y*b.y + b.z*b.z + b.w*b.w;
            if (q + 1 < Q_)  // warm L2/L0 for the next codebook (global_prefetch_b8)
                __builtin_prefetch(cbg + (size_t)K_ * D_ + r * D_, 0, 1);
        }
        __syncthreads();

        const float sc = post_scale[q];
        const float bi = post_bias[q];

        // A = -2 * residual (f32 WMMA has no A-negate modifier)
        v2f A1; A1.x = -2.f * r0; A1.y = -2.f * r1;
        v2f A2; A2.x = -2.f * r2; A2.y = -2.f * r3;

        float best[8]; int bestk[8];
        #pragma unroll
        for (int m = 0; m < 8; ++m) { best[m] = 3.4e38f; bestk[m] = 0; }

        #pragma unroll 2
        for (int nt = 0; nt < K_ / 16; ++nt) {
            const int n = (nt << 4) + tl;          // this lane's column
            const float* row = &cb_sh[n * CBS];
            v2f B1; B1.x = row[hl];     B1.y = row[2 + hl];   // K=0..3 rows
            v2f B2; B2.x = row[4 + hl]; B2.y = row[6 + hl];   // K=4..7 rows
            v8f c = {};                            // inline-0 C matrix
            c = __builtin_amdgcn_wmma_f32_16x16x4_f32(false, A1, false, B1,
                                                      (short)0, c, false, false);
            c = __builtin_amdgcn_wmma_f32_16x16x4_f32(false, A2, false, B2,
                                                      (short)0, c, false, false);
            const float cn = cn_sh[n];             // |c_n|^2, off the WMMA path
            #pragma unroll
            for (int m = 0; m < 8; ++m) {
                const float s = c[m] + cn;
                if (s < best[m]) { best[m] = s; bestk[m] = n; }
            }
        }

        // Branchless cross-column argmin: pack (sortable value, column) into
        // u64 keys; min over keys == argmin with first-index tie-break.
        unsigned long long key[8];
        #pragma unroll
        for (int m = 0; m < 8; ++m)
            key[m] = ((unsigned long long)sortable(best[m]) << 32)
                   | (unsigned)bestk[m];
        #pragma unroll
        for (int m = 0; m < 8; ++m) {
            unsigned long long k = key[m];
            #pragma unroll
            for (int mk = 1; mk <= 8; mk <<= 1) {
                const unsigned lo = (unsigned)shflx_i((int)(unsigned)k, mk, lane);
                const unsigned hi = (unsigned)shflx_i((int)(unsigned)(k >> 32), mk, lane);
                const unsigned long long ok = ((unsigned long long)hi << 32) | lo;
                k = (ok < k) ? ok : k;
            }
            key[m] = k;
        }
        // Lane selects the register holding row hl*8 + (tl&7); one bpermute
        // then delivers each lane its own token's winning index.
        int sel = (int)(unsigned)key[0];
        #pragma unroll
        for (int m = 1; m < 8; ++m)
            sel = ((tl & 7) == m) ? (int)(unsigned)key[m] : sel;
        const int src = ((tl >> 3) << 4) | (tl & 7);   // lane owning row `tl`
        const int mykk = __builtin_amdgcn_ds_bpermute(src << 2, sel) & (K_ - 1);

        // Gather embed for this lane's 4 dims; update residual/qsum/loss.
        const float* er = &cb_sh[mykk * CBS];
        const float e0 = er[c0 + 0], e1 = er[c0 + 1];
        const float e2 = er[c0 + 4], e3 = er[c0 + 5];
        const float d0 = r0 - 2.f * e0, d1 = r1 - 2.f * e1;
        const float d2 = r2 - 2.f * e2, d3 = r3 - 2.f * e3;
        lacc += d0 * d0 + d1 * d1 + d2 * d2 + d3 * d3;   // (r - 2e)^2
        q0 += e0 * sc + bi; q1 += e1 * sc + bi;
        q2 += e2 * sc + bi; q3 += e3 * sc + bi;
        r0 -= e0; r1 -= e1; r2 -= e2; r3 -= e3;

        if (hl == 0) out_codes[(size_t)q * BT_ + tok] = mykk;
    }

    // ---- final 1x1 conv: out[tok][e] = sum_d qsum[d] * w[e][d] + b[e] ----
    __syncthreads();                      // done reading cb_sh; reuse as stage
    const int tli = wave * TPW + tl;      // 0..127
    {
        float* qs = &cb_sh[tli * 9];      // stride 9: conflict-free
        qs[c0 + 0] = q0; qs[c0 + 1] = q1; qs[c0 + 4] = q2; qs[c0 + 5] = q3;
    }
    __syncthreads();
    {
        const float* qs = &cb_sh[tli * 9];
        float in[8];
        #pragma unroll
        for (int d = 0; d < 8; ++d) in[d] = qs[d];
        const int e0 = hl * 4;            // each half-wave makes 4 of 8 outputs
        float4 o;
        float* op = &o.x;
        #pragma unroll
        for (int e = 0; e < 4; ++e) {
            const float* w = conv_w + (size_t)(e0 + e) * D_;
            float acc = conv_b[e0 + e];
            #pragma unroll
            for (int d = 0; d < 8; ++d) acc += in[d] * w[d];
            op[e] = acc;
        }
        ((float4*)(out_q + (size_t)tok * D_))[hl] = o;
    }

    // ---- deterministic loss partial per block ----
    #pragma unroll
    for (int mk = 1; mk <= 16; mk <<= 1) lacc += shflx_f(lacc, mk, lane);
    if (lane == 0) ls_sh[wave] = lacc;
    __syncthreads();
    if (tid == 0) {
        float s = 0.f;
        #pragma unroll
        for (int w = 0; w < WAVES; ++w) s += ls_sh[w];
        loss_partials[blockIdx.x] = s;
    }
}

__global__ __launch_bounds__(256) void rvq_loss_reduce(
    const float* __restrict__ partials, float* __restrict__ loss_out, int n)
{
    __shared__ float sm[256];
    float s = 0.f;
    for (int i = threadIdx.x; i < n; i += 256) s += partials[i];  // fixed order
    sm[threadIdx.x] = s;
    __syncthreads();
    for (int off = 128; off > 0; off >>= 1) {
        if (threadIdx.x < (unsigned)off) sm[threadIdx.x] += sm[threadIdx.x + off];
        __syncthreads();
    }
    if (threadIdx.x == 0)
        *loss_out = sm[0] * (0.25f / (float)((size_t)BT_ * D_));
}

extern "C" void kernel_launch(void* const* d_in, const int* in_sizes, int n_in,
                              void* d_out, int out_size, void* d_ws, size_t ws_size,
                              hipStream_t stream) {
    const float* x          = (const float*)d_in[0];
    const float* codebooks  = (const float*)d_in[1];
    const float* post_scale = (const float*)d_in[2];
    const float* post_bias  = (const float*)d_in[3];
    const float* conv_w     = (const float*)d_in[4];
    const float* conv_b     = (const float*)d_in[5];

    // d_out layout (return order, flat): quantized [B,T,D] f32, loss scalar,
    // codes [Q,B,T] int32 (stored as raw 32-bit values).
    float* out_q    = (float*)d_out;
    float* out_loss = out_q + (size_t)BT_ * D_;
    int*   out_codes = (int*)(out_q + (size_t)BT_ * D_ + 1);

    float* partials = (float*)d_ws;  // NWG floats

    rvq_wmma_kernel<<<NWG, 256, 0, stream>>>(x, codebooks, post_scale, post_bias,
                                             conv_w, conv_b, out_q, out_codes,
                                             partials);
    rvq_loss_reduce<<<1, 256, 0, stream>>>(partials, out_loss, NWG);
}